// LJLKWholePoseScoringModule_5574867550317
// MI455X (gfx1250) — compile-verified
//
#include <hip/hip_runtime.h>
#include <math.h>

#define B_      48
#define A_      24
#define C_      4
#define NPAIR   (B_ * (B_ + 1) / 2)   // 1176 upper-tri block pairs (incl. diagonal)
#define INV2PI32 0.08979350102700918f

// spline / branch constants (folded at compile time)
#define INV45    0.2222222222222222f     // 1/4.5
#define INV45_2  0.04938271604938271f    // 1/4.5^2
#define INV45_3  0.010973936899862826f   // 1/4.5^3
#define C6_LIN   21.433470507544581f     // (1/0.6)^6
#define C12_LIN  459.39364480302213f     // (1/0.6)^12
#define F0_LIN   (C12_LIN - 2.0f * C6_LIN)          // lj_f(0.6*sigma)/eps
#define DF0_LIN  (-20.0f * (C12_LIN - C6_LIN))      // lj_df(0.6*sigma)*sigma/eps

typedef __attribute__((ext_vector_type(2))) float v2f;
typedef __attribute__((ext_vector_type(8))) float v8f;

#if defined(__has_builtin)
#if __has_builtin(__builtin_amdgcn_wmma_f32_16x16x4_f32)
#define HAVE_WMMA4 1
#endif
#endif
#ifndef HAVE_WMMA4
#define HAVE_WMMA4 0
#endif

__device__ __forceinline__ float hermite_to_zero(float d, float f0, float df0) {
    // cubic hermite on [4.5, 6.0] fading to (6.0, 0, 0)
    float t = (d - 4.5f) * (1.0f / 1.5f);
    t = fminf(fmaxf(t, 0.0f), 1.0f);
    float omt = 1.0f - t;
    return (f0 * (1.0f + 2.0f * t) + df0 * 1.5f * t) * omt * omt;
}

__global__ void ljlk_zero_kernel(float* out, int n) {
    int i = threadIdx.x;
    if (i < n) out[i] = 0.0f;
}

__global__ __launch_bounds__(128) void ljlk_pose_kernel(
    const float* __restrict__ coords,              // [P, B*A, 3]
    const int*   __restrict__ block_coord_offset,  // [P, B]
    const int*   __restrict__ block_types,         // [P, B]
    const int*   __restrict__ inter_block_bondsep, // [P, B, B, C, C]
    const int*   __restrict__ bt_n_atoms,          // [NT]
    const int*   __restrict__ bt_atom_types,       // [NT, A]
    const int*   __restrict__ bt_atom_is_heavy,    // [NT, A]
    const int*   __restrict__ bt_conn_atoms,       // [NT, C]
    const int*   __restrict__ bt_path_distance,    // [NT, A, A]
    const float* __restrict__ tp,                  // [T, 9]
    const float* __restrict__ gp,                  // [3]
    float* __restrict__ out,                       // [2, P] flat
    int Pn)
{
    __shared__ float sx[2][32], sy[2][32], sz[2][32], snorm[2][32];
    __shared__ float srad[2][A_], swdep[2][A_];
    __shared__ float silam[2][A_];    // 1/lambda
    __shared__ float sglam[2][A_];    // -dgfree/lambda
    __shared__ float svol[2][A_];
    __shared__ int   sflags[2][A_];
    __shared__ int   sd2c[2][C_][A_]; // path dist from conn atom c to atom a
    __shared__ int   stmp[C_][A_];    // min over c2 of ibb[c1][c2] + d2c2[c2][j]
    __shared__ float sred[2];

    const int tid = threadIdx.x;
    const int blk = blockIdx.x;
    const int p   = blk / NPAIR;
    int q = blk - p * NPAIR;

    // map q -> (b1 <= b2)
    int b1 = 0;
    for (int r = 0; r < B_; ++r) {
        int cnt = B_ - r;
        if (q < cnt) { b1 = r; break; }
        q -= cnt;
    }
    const int b2    = b1 + q;
    const bool diag = (b1 == b2);

    const int bt1  = block_types[p * B_ + b1];
    const int bt2  = block_types[p * B_ + b2];
    const int off1 = block_coord_offset[p * B_ + b1];
    const int off2 = block_coord_offset[p * B_ + b2];

    const float g0 = gp[0], g1 = gp[1], g2 = gp[2];

    // ---- phase 1: stage atom data for both blocks into LDS ----
    if (tid < 48) {
        int bsel = tid / A_;
        int a    = tid - bsel * A_;
        int bt   = bsel ? bt2 : bt1;
        int off  = bsel ? off2 : off1;
        int at   = bt_atom_types[bt * A_ + a];
        const float* cb = coords + ((size_t)p * (B_ * A_) + off + a) * 3;
        float x = cb[0], y = cb[1], z = cb[2];
        sx[bsel][a] = x; sy[bsel][a] = y; sz[bsel][a] = z;
        snorm[bsel][a] = x * x + y * y + z * z;
        const float* t9 = tp + at * 9;
        float il = 1.0f / t9[3];          // once per atom: precise divide is fine
        srad[bsel][a]  = t9[0];
        swdep[bsel][a] = t9[1];
        silam[bsel][a] = il;
        sglam[bsel][a] = -t9[2] * il;
        svol[bsel][a]  = t9[4];
        int fl = 0;
        if (t9[5] > 0.5f) fl |= 1;   // donor
        if (t9[6] > 0.5f) fl |= 2;   // hydroxyl
        if (t9[7] > 0.5f) fl |= 4;   // polar H
        if (t9[8] > 0.5f) fl |= 8;   // acceptor
        if (bt_atom_is_heavy[bt * A_ + a]) fl |= 16;
        if (a < bt_n_atoms[bt])            fl |= 32;
        sflags[bsel][a] = fl;
    } else if (tid < 64) {
        int u = tid - 48;
        int bsel = u >> 3;
        int a = A_ + (u & 7);        // pad entries 24..31 so WMMA inputs are defined
        sx[bsel][a] = 0.0f; sy[bsel][a] = 0.0f; sz[bsel][a] = 0.0f; snorm[bsel][a] = 0.0f;
    }
    for (int idx = tid; idx < 2 * C_ * A_; idx += 128) {
        int bsel = idx / (C_ * A_);
        int rem  = idx - bsel * (C_ * A_);
        int c = rem / A_, a = rem - (rem / A_) * A_;
        int bt = bsel ? bt2 : bt1;
        int conn = bt_conn_atoms[bt * C_ + c];
        sd2c[bsel][c][a] = bt_path_distance[(bt * A_ + conn) * A_ + a];
    }
    if (tid < 2) sred[tid] = 0.0f;
    __syncthreads();

    // ---- phase 2: min-plus over second connection ----
    if (tid < C_ * A_) {
        int c1 = tid / A_, j = tid - (tid / A_) * A_;
        const int* ibb = inter_block_bondsep + ((((size_t)p * B_ + b1) * B_ + b2) * C_ + c1) * C_;
        int m = 99;
        #pragma unroll
        for (int c2 = 0; c2 < C_; ++c2) m = min(m, ibb[c2] + sd2c[1][c2][j]);
        stmp[c1][j] = m;
    }
    __syncthreads();

    // ---- WMMA: 16x16 tile of squared distances per wave ----
    const int wave = tid >> 5;
    const int lane = tid & 31;
    const int half = lane >> 4;
    const int l    = lane & 15;
    const int tbm  = (wave >> 1) << 4;   // row-tile base (block1 atoms)
    const int tbn  = (wave & 1) << 4;    // col-tile base (block2 atoms)

#if HAVE_WMMA4
    // A (16x4): row m = (-2x, -2y, -2z, 1); lanes 0-15 K=0,1; lanes 16-31 K=2,3
    float xm = sx[0][tbm + l], ym = sy[0][tbm + l], zm = sz[0][tbm + l];
    v2f avA;
    avA.x = half ? (-2.0f * zm) : (-2.0f * xm);
    avA.y = half ? 1.0f         : (-2.0f * ym);
    // B (4x16): col n = (x, y, z, |p|^2); VGPR r holds K=r (lanes 0-15) / K=r+2 (lanes 16-31)
    float xn = sx[1][tbn + l], yn = sy[1][tbn + l], zn = sz[1][tbn + l], nn = snorm[1][tbn + l];
    v2f avB;
    avB.x = half ? zn : xn;
    avB.y = half ? nn : yn;
    // C: |p1_m|^2 per the 16x16 f32 C/D layout (VGPR r -> M = r + half*8, lane -> N)
    v8f avC;
    #pragma unroll
    for (int r = 0; r < 8; ++r) avC[r] = snorm[0][tbm + half * 8 + r];
    v8f D = __builtin_amdgcn_wmma_f32_16x16x4_f32(
        false, avA, false, avB, (short)0, avC, false, false);
#endif

    // lane-private column (block2 atom) parameters
    const int j = tbn + l;
    const bool jvalid = (j < A_);
    int fj = 0; float rj = 1.0f, wj = 0.0f, cj = 0.0f, ilj = 1.0f, vj = 0.0f;
    int tmpc[C_] = {99, 99, 99, 99};
    if (jvalid) {
        fj  = sflags[1][j];
        rj  = srad[1][j];  wj = swdep[1][j];
        cj  = sglam[1][j]; ilj = silam[1][j]; vj = svol[1][j];
        #pragma unroll
        for (int c = 0; c < C_; ++c) tmpc[c] = stmp[c][j];
    }

    float ljacc = 0.0f, lkacc = 0.0f;
    #pragma unroll
    for (int r = 0; r < 8; ++r) {
        int i = tbm + half * 8 + r;
        if (i >= A_ || !jvalid) continue;
        if (diag && i >= j) continue;              // count each intra-block pair once
        int fi = sflags[0][i];
        if (!((fi & 32) && (fj & 32))) continue;   // both atoms valid

#if HAVE_WMMA4
        float d2v = D[r];
#else
        float dx = sx[0][i] - sx[1][j];
        float dy = sy[0][i] - sy[1][j];
        float dz = sz[0][i] - sz[1][j];
        float d2v = dx * dx + dy * dy + dz * dz;
#endif
        d2v = fmaxf(d2v, 1e-6f);
        if (d2v >= 36.0f) continue;                // both terms zero past 6 A
        float invd = __builtin_amdgcn_rsqf(d2v);   // v_rsq_f32
        float d    = d2v * invd;

        // bond separation / count-pair weight
        int sep;
        if (diag) {
            sep = bt_path_distance[(bt1 * A_ + i) * A_ + j];
        } else {
            sep = 99;
            #pragma unroll
            for (int c = 0; c < C_; ++c) sep = min(sep, sd2c[0][c][i] + tmpc[c]);
        }
        if (sep < 4) continue;
        float w = (sep > 4) ? 1.0f : 0.2f;

        float ri = srad[0][i], wi = swdep[0][i];

        // sigma with donor/acceptor overrides (same priority order as reference)
        float sigma = ri + rj;
        bool di_ = fi & 1, hi_ = fi & 2, pi_ = fi & 4, ai_ = fi & 8;
        bool dj_ = fj & 1, hj_ = fj & 2, pj_ = fj & 4, aj_ = fj & 8;
        if ((pi_ && aj_) || (pj_ && ai_)) sigma = g2;
        if ((di_ && hi_ && aj_) || (dj_ && hj_ && ai_)) sigma = g1;
        if ((di_ && !hi_ && aj_) || (dj_ && !hj_ && ai_)) sigma = g0;
        float eps = __builtin_amdgcn_sqrtf(wi * wj);

        // ---- LJ: linear < 0.6*sigma, 12-6 mid, hermite fade [4.5, 6] ----
        float lj;
        float dlin = 0.6f * sigma;
        if (d < dlin) {
            // sigma/dlin is the constant 1/0.6 -> fully folded anchors
            float rsig = __builtin_amdgcn_rcpf(sigma);
            lj = eps * (F0_LIN + DF0_LIN * rsig * (d - dlin));
        } else if (d < 4.5f) {
            float s = sigma * invd, s2 = s * s, s6 = s2 * s2 * s2;
            lj = eps * (s6 * s6 - 2.0f * s6);
        } else {
            float s = sigma * INV45, s2 = s * s, s6 = s2 * s2 * s2;
            float f0  = eps * (s6 * s6 - 2.0f * s6);
            float df0 = eps * (-12.0f * INV45) * (s6 * s6 - s6);
            lj = hermite_to_zero(d, f0, df0);
        }
        ljacc += w * lj;

        // ---- LK desolvation (heavy-heavy only) ----
        if ((fi & 16) && (fj & 16)) {
            float ci = sglam[0][i], ili = silam[0][i], vi = svol[0][i];
            float dnear = ri + rj;
            float lk;
            if (d < dnear) {
                // dnear - ri == rj, dnear - rj == ri
                float xi = rj * ili, xj = ri * ilj;
                float ti = ci * vj * __expf(-xi * xi);
                float tj = cj * vi * __expf(-xj * xj);
                float invdn = __builtin_amdgcn_rcpf(dnear);
                lk = INV2PI32 * (ti + tj) * invdn * invdn;
            } else if (d < 4.5f) {
                float xi = (d - ri) * ili, xj = (d - rj) * ilj;
                float ti = ci * vj * __expf(-xi * xi);
                float tj = cj * vi * __expf(-xj * xj);
                lk = INV2PI32 * (ti + tj) * invd * invd;
            } else {
                float xi = (4.5f - ri) * ili, xj = (4.5f - rj) * ilj;
                float ti = ci * vj * __expf(-xi * xi);
                float tj = cj * vi * __expf(-xj * xj);
                float f0  = INV2PI32 * (ti + tj) * INV45_2;
                float df0 = INV2PI32 * ((ti * (-2.0f * xi * ili) + tj * (-2.0f * xj * ilj)) * INV45_2
                                        - 2.0f * (ti + tj) * INV45_3);
                lk = hermite_to_zero(d, f0, df0);
            }
            lkacc += w * lk;
        }
    }

    // ---- reduction: wave32 shuffle tree -> LDS -> 2 global atomics ----
    #pragma unroll
    for (int off = 16; off > 0; off >>= 1) {
        ljacc += __shfl_xor(ljacc, off, 32);
        lkacc += __shfl_xor(lkacc, off, 32);
    }
    if (lane == 0) {
        atomicAdd(&sred[0], ljacc);
        atomicAdd(&sred[1], lkacc);
    }
    __syncthreads();
    if (tid == 0) {
        atomicAdd(&out[p],      sred[0]);
        atomicAdd(&out[Pn + p], sred[1]);
    }
}

extern "C" void kernel_launch(void* const* d_in, const int* in_sizes, int n_in,
                              void* d_out, int out_size, void* d_ws, size_t ws_size,
                              hipStream_t stream) {
    const float* coords  = (const float*)d_in[0];
    const int*   bco     = (const int*)d_in[1];
    const int*   btv     = (const int*)d_in[2];
    // d_in[3] = min_block_bondsep (early-exit hint, unused; dense evaluation)
    const int*   ibb     = (const int*)d_in[4];
    const int*   nat     = (const int*)d_in[5];
    const int*   atyp    = (const int*)d_in[6];
    const int*   aheavy  = (const int*)d_in[7];
    const int*   conn    = (const int*)d_in[8];
    const int*   pd      = (const int*)d_in[9];
    const float* tp      = (const float*)d_in[10];
    const float* gp      = (const float*)d_in[11];
    float*       out     = (float*)d_out;

    const int Pn = in_sizes[2] / B_;   // block_types is [P, B]

    hipLaunchKernelGGL(ljlk_zero_kernel, dim3(1), dim3(64), 0, stream, out, out_size);
    hipLaunchKernelGGL(ljlk_pose_kernel, dim3(Pn * NPAIR), dim3(128), 0, stream,
                       coords, bco, btv, ibb, nat, atyp, aheavy, conn, pd, tp, gp, out, Pn);
}